// CRFLayer_1374389534847
// MI455X (gfx1250) — compile-verified
//
#include <hip/hip_runtime.h>
#include <cstdint>
#include <cstddef>

#define L 32            // tag count (compile-time; matches reference)
#define CH 64           // timesteps per LDS chunk: 64*32*4B = 8 KB per buffer
#define MASK_ID 0

typedef unsigned int  v4u __attribute__((ext_vector_type(4)));
typedef int           v8i __attribute__((ext_vector_type(8)));
typedef int           v4i __attribute__((ext_vector_type(4)));

// ---------- CDNA5 async data movement: prefer TDM, then async-to-LDS, then sync ----------
#if __has_builtin(__builtin_amdgcn_tensor_load_to_lds)
  #define USE_TDM 1
  #define COPIES_PER_CHUNK 1
  #define WAIT_COPY(N) __builtin_amdgcn_s_wait_tensorcnt(N)
#elif __has_builtin(__builtin_amdgcn_global_load_async_to_lds_b128)
  #define USE_ASYNC 1
  #define COPIES_PER_CHUNK (CH * L * 4 / 512)   // 16 b128 ops x 32 lanes per 8KB chunk
  #if __has_builtin(__builtin_amdgcn_s_wait_asynccnt)
    #define WAIT_COPY(N) __builtin_amdgcn_s_wait_asynccnt(N)
  #else
    #define WAIT_COPY(N) asm volatile("s_wait_asynccnt %0" ::"i"(N))
  #endif
#else
  #define USE_SYNC 1
  #define COPIES_PER_CHUNK 0
  #define WAIT_COPY(N)
#endif

#if defined(USE_ASYNC)
typedef const char __attribute__((address_space(1)))* gas1_t;
typedef char __attribute__((address_space(3)))*       las3_t;
#endif

// Copy CH*L floats (8 KB) from global to LDS, asynchronously where possible.
__device__ __forceinline__ void chunk_copy(float* lbuf, const float* gsrc, int lane) {
#if defined(USE_TDM)
  // Build a Tensor DMA Descriptor (D#): 1 row of CH*L fp32 elements.
  unsigned lds = (unsigned)(uintptr_t)lbuf;                 // flat->LDS offset (low 32 bits)
  unsigned long long ga = (unsigned long long)(uintptr_t)gsrc;
  v4u g0;
  g0[0] = 1u;                                               // count=1, is_restore=0, gather off
  g0[1] = lds;                                              // lds_addr [63:32]
  g0[2] = (unsigned)ga;                                     // global_addr[31:0]
  g0[3] = (unsigned)((ga >> 32) & 0x1FFFFFFull)             // global_addr[56:32] -> bits[120:96]
        | (2u << 30);                                       // type=2 ("image") bits[127:126]
  v8i g1;
  g1[0] = (int)(2u << 16);                                  // workgroup_mask=0, data_size=2 (4B)
  g1[1] = 0;                                                // tensor_dim0[15:0]=0 (dim0=0x40000000)
  g1[2] = (int)(0x4000u | (1u << 16));                      // tensor_dim0[31:16]; tensor_dim1 lo16 = 1
  g1[3] = (int)((unsigned)(CH * L) << 16);                  // tensor_dim1 hi16 = 0; tile_dim0 = CH*L
  g1[4] = 1;                                                // tile_dim1 = 1; tile_dim2 = 0
  g1[5] = (int)(CH * L);                                    // tensor_dim0_stride lo32
  g1[6] = 0;                                                // stride hi / dim1_stride lo
  g1[7] = 0;
  v4i z4 = {0, 0, 0, 0};
  v8i z8 = {0, 0, 0, 0, 0, 0, 0, 0};
  __builtin_amdgcn_tensor_load_to_lds(g0, g1, z4, z4, z8, 0);
  (void)lane;
#elif defined(USE_ASYNC)
  gas1_t g = (gas1_t)(uintptr_t)gsrc;
  las3_t l = (las3_t)(unsigned)(uintptr_t)lbuf;
  #pragma unroll
  for (int k = 0; k < COPIES_PER_CHUNK; ++k) {
    __builtin_amdgcn_global_load_async_to_lds_b128(
        g + (size_t)lane * 16 + (size_t)k * 512,
        l + lane * 16 + k * 512, 0, 0);
  }
#else
  // Synchronous fallback: vectorized 16B copies (wave in-order LDS keeps this safe).
  #pragma unroll
  for (int k = lane; k < (CH * L) / 4; k += 32)
    ((float4*)lbuf)[k] = ((const float4*)gsrc)[k];
#endif
}

__device__ __forceinline__ float bcast_f(float v, int i) {
#if __has_builtin(__builtin_amdgcn_readlane)
  return __int_as_float(__builtin_amdgcn_readlane(__float_as_int(v), i));
#else
  return __shfl(v, i, 32);
#endif
}

__global__ __launch_bounds__(32)
void viterbi_crf_kernel(const float* __restrict__ emis,      // [B,T,L]
                        const int*   __restrict__ seqlen,    // [B]
                        const float* __restrict__ trans,     // [L,L]
                        int*         __restrict__ out,       // [B,T] int32
                        unsigned char* __restrict__ bpws,    // [B,T,L] backpointers
                        int T)
{
  __shared__ __align__(16) float buf[2][CH * L];   // 16 KB double buffer
  const int lane = threadIdx.x;                    // 0..31 == tag id
  const int b = blockIdx.x;

  int n = seqlen[b];
  n = (n < 1) ? 1 : ((n > T) ? T : n);

  // Lane j holds transition column trans[:, j] in 32 VGPRs.
  float trc[L];
  #pragma unroll
  for (int i = 0; i < L; ++i) trc[i] = trans[i * L + lane];

  const float*   src  = emis + (size_t)b * T * L;
  unsigned char* bpr  = bpws + (size_t)b * T * L;
  int*           orow = out  + (size_t)b * T;

  const int nChunks = (n + CH - 1) / CH;

  // ---- forward pass: double-buffered chunk streaming ----
  chunk_copy(&buf[0][0], src, lane);
  if (nChunks > 1) {
    chunk_copy(&buf[1][0], src + CH * L, lane);
    WAIT_COPY(COPIES_PER_CHUNK);    // chunk 0 ready (in-order completion)
  } else {
    WAIT_COPY(0);
  }

  float alpha = 0.0f;
  for (int c = 0; c < nChunks; ++c) {
    const float* lb = &buf[c & 1][0];
    const int t0 = c * CH;
    const int t1 = (n < t0 + CH) ? n : (t0 + CH);
    int ts = t0;
    if (c == 0) { alpha = lb[lane]; ts = 1; }     // alpha0 = emissions[b,0,:]
    for (int t = ts; t < t1; ++t) {
      const float logit = lb[(t - t0) * L + lane];
      // max-plus row: best over previous tags, first-max tie-break (strict >)
      float best = bcast_f(alpha, 0) + trc[0];
      int bi = 0;
      #pragma unroll
      for (int i = 1; i < L; ++i) {
        const float s = bcast_f(alpha, i) + trc[i];
        if (s > best) { best = s; bi = i; }
      }
      alpha = best + logit;
      bpr[(size_t)t * L + lane] = (unsigned char)bi;
    }
    if (c + 1 < nChunks) {
      if (c + 2 < nChunks) {
        chunk_copy(&buf[c & 1][0], src + (size_t)(c + 2) * CH * L, lane);
        WAIT_COPY(COPIES_PER_CHUNK);  // ensure chunk c+1 is ready
      } else {
        WAIT_COPY(0);
      }
    }
  }

  // ---- wave-wide argmax over alpha (first-max tie break) ----
  float v = alpha; int idx = lane;
  #pragma unroll
  for (int off = 16; off >= 1; off >>= 1) {
    float ov = __shfl_xor(v, off, 32);
    int   oi = __shfl_xor(idx, off, 32);
    if (ov > v || (ov == v && oi < idx)) { v = ov; idx = oi; }
  }
  int tag = idx;   // uniform across wave

  // ---- masked tail -> MASK_ID, coalesced ----
  for (int tt = n + lane; tt < T; tt += 32) orow[tt] = MASK_ID;

  if (lane == 0) orow[n - 1] = tag;

  // ---- backtrack: 16 bp rows per round, b128/lane, next round prefetched ----
  int t = n - 1;
  uint4 w = make_uint4(0u, 0u, 0u, 0u);
  {
    int r = t - (lane >> 1);                    // lanes 2k,2k+1 hold row (t - k)
    if (r >= 1)
      w = *(const uint4*)(bpr + (size_t)r * L + (size_t)(lane & 1) * 16);
  }
  while (t >= 1) {
    uint4 wn = make_uint4(0u, 0u, 0u, 0u);
    {
      int r = (t - 16) - (lane >> 1);
      if (r >= 1)
        wn = *(const uint4*)(bpr + (size_t)r * L + (size_t)(lane & 1) * 16);
    }
    const int kmax = (t < 16) ? t : 16;
    for (int k = 0; k < kmax; ++k) {
      const unsigned dsel = ((unsigned)tag >> 2) & 3u;            // uniform
      unsigned dw = (dsel == 0u) ? w.x : (dsel == 1u) ? w.y
                  : (dsel == 2u) ? w.z : w.w;
      const int srcLane = k * 2 + (tag >> 4);                     // uniform
      dw = (unsigned)__shfl((int)dw, srcLane, 32);
      tag = (int)((dw >> (((unsigned)tag & 3u) * 8u)) & 31u);
      if (lane == 0) orow[t - 1] = tag;
      --t;
    }
    w = wn;
  }
}

extern "C" void kernel_launch(void* const* d_in, const int* in_sizes, int n_in,
                              void* d_out, int out_size, void* d_ws, size_t ws_size,
                              hipStream_t stream) {
  (void)n_in; (void)out_size; (void)ws_size;
  const float* emis   = (const float*)d_in[0];   // [B,T,L] f32
  const int*   seqlen = (const int*)d_in[1];     // [B] i32
  const float* trans  = (const float*)d_in[2];   // [L,L] f32
  const int B = in_sizes[1];
  const int T = in_sizes[0] / (B * L);
  viterbi_crf_kernel<<<dim3(B), dim3(32), 0, stream>>>(
      emis, seqlen, trans, (int*)d_out, (unsigned char*)d_ws, T);
}